// Net_21646635172354
// MI455X (gfx1250) — compile-verified
//
#include <hip/hip_runtime.h>
#include <cstdint>
#include <math.h>

#define N_NODES 6890
#define N_EDGES 41340
#define NPAIR   (N_EDGES * 8)
#define KTAB    125

typedef float v2f __attribute__((ext_vector_type(2)));
typedef float v8f __attribute__((ext_vector_type(8)));
typedef unsigned int u32x4 __attribute__((ext_vector_type(4)));
typedef int i32x4 __attribute__((ext_vector_type(4)));
typedef int i32x8 __attribute__((ext_vector_type(8)));

__device__ __forceinline__ float elu1(float x) { return x > 0.f ? x : __expf(x) - 1.f; }

// ---------------- basis + counting sort prep ----------------
__global__ void basis_kernel(const float* __restrict__ pseudo, const int* __restrict__ ei,
                             int* __restrict__ wiA, float* __restrict__ bA,
                             int* __restrict__ cnt, int* __restrict__ degI) {
  int e = blockIdx.x * blockDim.x + threadIdx.x;
  if (e >= N_EDGES) return;
  float fr[3]; int k0[3];
#pragma unroll
  for (int d = 0; d < 3; d++) {
    float p = pseudo[e * 3 + d] * 4.0f;
    float fl = floorf(p);
    fl = fminf(fmaxf(fl, 0.f), 3.f);
    k0[d] = (int)fl;
    fr[d] = p - fl;
  }
#pragma unroll
  for (int s = 0; s < 8; s++) {
    int b0 = s & 1, b1 = (s >> 1) & 1, b2 = (s >> 2) & 1;
    float w = (b0 ? fr[0] : 1.f - fr[0]) * (b1 ? fr[1] : 1.f - fr[1]) * (b2 ? fr[2] : 1.f - fr[2]);
    int wi = (k0[0] + b0) + 5 * (k0[1] + b1) + 25 * (k0[2] + b2);
    wiA[e * 8 + s] = wi;
    bA[e * 8 + s] = w;
    atomicAdd(&cnt[wi], 1);
  }
  atomicAdd(&degI[ei[e]], 1);  // out-degree from src, as in reference
}

__global__ void scan_kernel(const int* __restrict__ cnt, int* __restrict__ off, int* __restrict__ cursor) {
  if (threadIdx.x == 0 && blockIdx.x == 0) {
    int acc = 0;
    for (int k = 0; k < KTAB; k++) { off[k] = acc; cursor[k] = acc; acc += cnt[k]; }
    off[KTAB] = acc;
  }
}

__global__ void scatter_kernel(const int* __restrict__ wiA, const float* __restrict__ bA,
                               int* __restrict__ pe, float* __restrict__ pb, int* __restrict__ cursor) {
  int pId = blockIdx.x * blockDim.x + threadIdx.x;
  if (pId >= NPAIR) return;
  int k = wiA[pId];
  int pos = atomicAdd(&cursor[k], 1);
  pe[pos] = pId >> 3;    // edge id
  pb[pos] = bA[pId];     // basis weight
}

__global__ void dinv_kernel(const int* __restrict__ degI, float* __restrict__ dinv) {
  int n = blockIdx.x * blockDim.x + threadIdx.x;
  if (n < N_NODES) dinv[n] = 1.0f / ((float)degI[n] + 1.0f);
}

// ---------------- layer 0 (Fin=1, Fout=32) ----------------
__global__ void l0_agg_kernel(const float* __restrict__ x, const int* __restrict__ ei,
                              const int* __restrict__ wiA, const float* __restrict__ bA,
                              const float* __restrict__ W0, float* __restrict__ agg) {
  int t = blockIdx.x * blockDim.x + threadIdx.x;
  if (t >= N_EDGES * 32) return;
  int e = t >> 5, o = t & 31;
  float acc = 0.f;
#pragma unroll
  for (int s = 0; s < 8; s++)
    acc += bA[e * 8 + s] * W0[wiA[e * 8 + s] * 32 + o];
  acc *= x[ei[e]];
  atomicAdd(&agg[ei[N_EDGES + e] * 32 + o], acc);
}

// ---------------- root + bias + degree-norm + elu ----------------
__global__ void finalize_kernel(const float* __restrict__ agg, const float* __restrict__ hin,
                                const float* __restrict__ root, const float* __restrict__ bias,
                                const float* __restrict__ dinv, float* __restrict__ hout, int Fin) {
  int n = blockIdx.x;
  int o = threadIdx.x;
  int Fout = blockDim.x;
  __shared__ float hrow[64];
  for (int f = o; f < Fin; f += Fout) hrow[f] = hin[n * Fin + f];
  __syncthreads();
  float acc = agg[n * Fout + o] + bias[o];
  for (int f = 0; f < Fin; f++) acc += hrow[f] * root[f * Fout + o];
  acc *= dinv[n];
  hout[n * Fout + o] = elu1(acc);
}

// ---------------- per-k gathered GEMM with f32 WMMA (Fout=64) ----------------
__global__ __launch_bounds__(32)
void spline_agg_kernel(const float* __restrict__ h, const int* __restrict__ ei,
                       const int* __restrict__ pe, const float* __restrict__ pb,
                       const int* __restrict__ off, const float* __restrict__ W,
                       float* __restrict__ agg, int Fin) {
  const int Fout = 64;
  int k = blockIdx.x;
  int start = off[k], end = off[k + 1];
  int cnt = end - start;
  if (cnt <= 0) return;  // uniform over the wave
  int lane = threadIdx.x;
  int row = lane & 15, half = lane >> 4;
  __shared__ float A[16][68];     // stride 68 -> conflict-free column reads
  __shared__ int dsts[16];
  const float* Wk = W + (size_t)k * Fin * Fout;
  int ntiles = (cnt + 15) >> 4;
  for (int t = blockIdx.y; t < ntiles; t += gridDim.y) {
    int p0 = start + t * 16;
    {  // gather 16 basis-prescaled source rows (2 lanes per row)
      int r = lane >> 1, hf = lane & 1;
      int pidx = p0 + r;
      float b = 0.f; int srcn = 0;
      if (pidx < end) {
        b = pb[pidx];
        int e = pe[pidx];
        srcn = ei[e];
        if (!hf) dsts[r] = ei[N_EDGES + e];
      } else if (!hf) dsts[r] = 0;
      int fh = Fin >> 1;
      const float* hs = h + (size_t)srcn * Fin + hf * fh;
      float* Ar = &A[r][hf * fh];
      for (int f = 0; f < fh; f++) Ar[f] = b * hs[f];
    }
    __syncthreads();
    v8f c0 = {}, c1 = {}, c2 = {}, c3 = {};
    for (int kk = 0; kk < Fin; kk += 4) {
      v2f a; a.x = A[row][kk + half * 2]; a.y = A[row][kk + half * 2 + 1];
      const float* B0 = Wk + (size_t)(kk + half * 2) * Fout;
      v2f b0, b1, b2, b3;
      b0.x = B0[row];      b0.y = B0[Fout + row];
      b1.x = B0[16 + row]; b1.y = B0[Fout + 16 + row];
      b2.x = B0[32 + row]; b2.y = B0[Fout + 32 + row];
      b3.x = B0[48 + row]; b3.y = B0[Fout + 48 + row];
      c0 = __builtin_amdgcn_wmma_f32_16x16x4_f32(false, a, false, b0, (short)0, c0, false, false);
      c1 = __builtin_amdgcn_wmma_f32_16x16x4_f32(false, a, false, b1, (short)0, c1, false, false);
      c2 = __builtin_amdgcn_wmma_f32_16x16x4_f32(false, a, false, b2, (short)0, c2, false, false);
      c3 = __builtin_amdgcn_wmma_f32_16x16x4_f32(false, a, false, b3, (short)0, c3, false, false);
    }
#pragma unroll
    for (int v = 0; v < 8; v++) {
      int M = v + half * 8;
      float* dp = agg + (size_t)dsts[M] * Fout;
      atomicAdd(dp + row,      c0[v]);
      atomicAdd(dp + 16 + row, c1[v]);
      atomicAdd(dp + 32 + row, c2[v]);
      atomicAdd(dp + 48 + row, c3[v]);
    }
    __syncthreads();
  }
}

// ---------------- fc1: [N,64]@[64,256] + bias + elu ----------------
__global__ __launch_bounds__(32)
void fc1_kernel(const float* __restrict__ hin, const float* __restrict__ Wf,
                const float* __restrict__ bf, float* __restrict__ hout) {
  const int Kd = 64, Nc = 256;
  int m0 = blockIdx.x * 16, n0 = blockIdx.y * 64;
  int lane = threadIdx.x, row = lane & 15, half = lane >> 4;
  __shared__ float A[16][68];
  for (int idx = lane; idx < 16 * Kd; idx += 32) {
    int r = idx >> 6, c = idx & 63;
    int gm = m0 + r;
    A[r][c] = (gm < N_NODES) ? hin[gm * Kd + c] : 0.f;
  }
  __syncthreads();
  v8f c0 = {}, c1 = {}, c2 = {}, c3 = {};
  for (int kk = 0; kk < Kd; kk += 4) {
    v2f a; a.x = A[row][kk + half * 2]; a.y = A[row][kk + half * 2 + 1];
    const float* B0 = Wf + (size_t)(kk + half * 2) * Nc + n0;
    v2f b0, b1, b2, b3;
    b0.x = B0[row];      b0.y = B0[Nc + row];
    b1.x = B0[16 + row]; b1.y = B0[Nc + 16 + row];
    b2.x = B0[32 + row]; b2.y = B0[Nc + 32 + row];
    b3.x = B0[48 + row]; b3.y = B0[Nc + 48 + row];
    c0 = __builtin_amdgcn_wmma_f32_16x16x4_f32(false, a, false, b0, (short)0, c0, false, false);
    c1 = __builtin_amdgcn_wmma_f32_16x16x4_f32(false, a, false, b1, (short)0, c1, false, false);
    c2 = __builtin_amdgcn_wmma_f32_16x16x4_f32(false, a, false, b2, (short)0, c2, false, false);
    c3 = __builtin_amdgcn_wmma_f32_16x16x4_f32(false, a, false, b3, (short)0, c3, false, false);
  }
#pragma unroll
  for (int v = 0; v < 8; v++) {
    int gm = m0 + v + half * 8;
    if (gm < N_NODES) {
      float* op = hout + (size_t)gm * Nc + n0;
      op[row]      = elu1(c0[v] + bf[n0 + row]);
      op[16 + row] = elu1(c1[v] + bf[n0 + 16 + row]);
      op[32 + row] = elu1(c2[v] + bf[n0 + 32 + row]);
      op[48 + row] = elu1(c3[v] + bf[n0 + 48 + row]);
    }
  }
}

// ---------------- fc2: [N,256]@[256,6890] + bias -> logits ----------------
// 8 waves per block; block tile 128(M) x 64(N); B panel staged to LDS by the
// Tensor Data Mover (wave 0), A chunks staged per wave. K chunked by 32.
__global__ __launch_bounds__(256)
void fc2_kernel(const float* __restrict__ hin, const float* __restrict__ Wf,
                const float* __restrict__ bf, float* __restrict__ out) {
  const int Kd = 256, Nc = N_NODES;
  const int BSTRIDE = 80;  // 64 cols + 16 pad dwords -> conflict-free frag reads
  int m0 = blockIdx.x * 128, n0 = blockIdx.y * 64;
  int wave = threadIdx.x >> 5;
  int lane = threadIdx.x & 31;
  int row = lane & 15, half = lane >> 4;
  int mw = m0 + wave * 16;
  __shared__ float Ash[8][16][36];       // per-wave A chunk, stride 36
  __shared__ float Bsh[32 * BSTRIDE];    // B chunk 32 x 64 (+pad)
  v8f c0 = {}, c1 = {}, c2 = {}, c3 = {};
  for (int kc = 0; kc < Kd; kc += 32) {
    __syncthreads();  // previous Bsh/Ash fully consumed
#if __has_builtin(__builtin_amdgcn_tensor_load_to_lds) && __has_builtin(__builtin_amdgcn_s_wait_tensorcnt)
    if (threadIdx.x < 32) {
      unsigned long long ga = (unsigned long long)(uintptr_t)(Wf + (size_t)kc * Nc + n0);
      unsigned int lb = (unsigned int)(uintptr_t)(&Bsh[0]);
      u32x4 g0; i32x8 g1;
      i32x4 g2 = {0, 0, 0, 0}, g3 = {0, 0, 0, 0};
      i32x8 g4 = {0, 0, 0, 0, 0, 0, 0, 0};
      // D# group0: count=1 | lds_addr | global_addr | type=2
      g0.x = 1u;
      g0.y = lb;
      g0.z = (unsigned int)ga;
      g0.w = (unsigned int)((ga >> 32) & 0x01FFFFFFu) | (2u << 30);
      int td0 = Nc - n0;   // remaining cols: OOB reads return 0
      int td1 = Kd - kc;   // remaining rows
      // D# group1: data_size=4B, pad_enable, pad_interval=64dw, pad_amount=16dw,
      //            tensor_dim0/1, tile 64x32, tensor_dim0_stride=Nc
      g1[0] = (2 << 16) | (1 << 20) | (5 << 22) | (15 << 25);
      g1[1] = (td0 & 0xFFFF) << 16;
      g1[2] = ((td0 >> 16) & 0xFFFF) | ((td1 & 0xFFFF) << 16);
      g1[3] = ((td1 >> 16) & 0xFFFF) | (64 << 16);
      g1[4] = 32;          // tile_dim1=32, tile_dim2=0
      g1[5] = Nc;          // tensor_dim0_stride (lo 32)
      g1[6] = 0; g1[7] = 0;
      __builtin_amdgcn_tensor_load_to_lds(g0, g1, g2, g3, g4, 0);
    }
#else
    for (int idx = threadIdx.x; idx < 32 * 64; idx += 256) {
      int r = idx >> 6, c = idx & 63;
      int col = n0 + c;
      Bsh[r * BSTRIDE + c] = (col < Nc) ? Wf[(size_t)(kc + r) * Nc + col] : 0.f;
    }
#endif
    // stage this wave's 16x32 A chunk while the TDM streams B
    for (int idx = lane; idx < 16 * 32; idx += 32) {
      int r = idx >> 5, c = idx & 31;
      int gm = mw + r;
      Ash[wave][r][c] = (gm < Nc) ? hin[(size_t)gm * Kd + kc + c] : 0.f;
    }
#if __has_builtin(__builtin_amdgcn_tensor_load_to_lds) && __has_builtin(__builtin_amdgcn_s_wait_tensorcnt)
    if (threadIdx.x < 32) __builtin_amdgcn_s_wait_tensorcnt(0);
#endif
    __syncthreads();
#pragma unroll
    for (int kk = 0; kk < 32; kk += 4) {
      v2f a; a.x = Ash[wave][row][kk + half * 2]; a.y = Ash[wave][row][kk + half * 2 + 1];
      const float* B0 = &Bsh[(kk + half * 2) * BSTRIDE];
      v2f b0, b1, b2, b3;
      b0.x = B0[row];      b0.y = B0[BSTRIDE + row];
      b1.x = B0[16 + row]; b1.y = B0[BSTRIDE + 16 + row];
      b2.x = B0[32 + row]; b2.y = B0[BSTRIDE + 32 + row];
      b3.x = B0[48 + row]; b3.y = B0[BSTRIDE + 48 + row];
      c0 = __builtin_amdgcn_wmma_f32_16x16x4_f32(false, a, false, b0, (short)0, c0, false, false);
      c1 = __builtin_amdgcn_wmma_f32_16x16x4_f32(false, a, false, b1, (short)0, c1, false, false);
      c2 = __builtin_amdgcn_wmma_f32_16x16x4_f32(false, a, false, b2, (short)0, c2, false, false);
      c3 = __builtin_amdgcn_wmma_f32_16x16x4_f32(false, a, false, b3, (short)0, c3, false, false);
    }
  }
  int cb = n0 + row;
#pragma unroll
  for (int v = 0; v < 8; v++) {
    int gm = mw + v + half * 8;
    if (gm < Nc) {
      float* op = out + (size_t)gm * Nc;
      if (cb      < Nc) op[cb]      = c0[v] + bf[cb];
      if (cb + 16 < Nc) op[cb + 16] = c1[v] + bf[cb + 16];
      if (cb + 32 < Nc) op[cb + 32] = c2[v] + bf[cb + 32];
      if (cb + 48 < Nc) op[cb + 48] = c3[v] + bf[cb + 48];
    }
  }
}

// ---------------- row-wise log_softmax, in place ----------------
__global__ void logsoftmax_kernel(float* __restrict__ out) {
  const int C = N_NODES;
  float* rowp = out + (size_t)blockIdx.x * C;
  __shared__ float red[256];
  int tid = threadIdx.x;
  float m = -3.402823466e38f;
  for (int c = tid; c < C; c += 256) m = fmaxf(m, rowp[c]);
  red[tid] = m; __syncthreads();
  for (int s = 128; s > 0; s >>= 1) { if (tid < s) red[tid] = fmaxf(red[tid], red[tid + s]); __syncthreads(); }
  m = red[0]; __syncthreads();
  float sum = 0.f;
  for (int c = tid; c < C; c += 256) sum += __expf(rowp[c] - m);
  red[tid] = sum; __syncthreads();
  for (int s = 128; s > 0; s >>= 1) { if (tid < s) red[tid] += red[tid + s]; __syncthreads(); }
  float ls = m + logf(red[0]);
  __syncthreads();
  for (int c = tid; c < C; c += 256) rowp[c] -= ls;
}

// ---------------- driver ----------------
extern "C" void kernel_launch(void* const* d_in, const int* in_sizes, int n_in,
                              void* d_out, int out_size, void* d_ws, size_t ws_size,
                              hipStream_t stream) {
  (void)in_sizes; (void)n_in; (void)out_size; (void)ws_size;
  const float* x      = (const float*)d_in[0];
  const int*   ei     = (const int*)d_in[1];
  const float* pseudo = (const float*)d_in[2];
  const float* Wc[6]; const float* Rc[6]; const float* Bc[6];
  for (int i = 0; i < 6; i++) {
    Wc[i] = (const float*)d_in[3 + i];
    Rc[i] = (const float*)d_in[9 + i];
    Bc[i] = (const float*)d_in[15 + i];
  }
  const float* fc1w = (const float*)d_in[21];
  const float* fc1b = (const float*)d_in[22];
  const float* fc2w = (const float*)d_in[23];
  const float* fc2b = (const float*)d_in[24];
  float* out = (float*)d_out;

  uint8_t* p = (uint8_t*)d_ws;
  auto alloc = [&](size_t bytes) -> void* {
    void* r = (void*)p; p += (bytes + 255) & ~(size_t)255; return r;
  };
  int*   cnt    = (int*)alloc(128 * 4);
  int*   off    = (int*)alloc(129 * 4);
  int*   cursor = (int*)alloc(128 * 4);
  int*   degI   = (int*)alloc(N_NODES * 4);
  float* dinv   = (float*)alloc(N_NODES * 4);
  int*   wiA    = (int*)alloc((size_t)NPAIR * 4);
  float* bA     = (float*)alloc((size_t)NPAIR * 4);
  int*   pe     = (int*)alloc((size_t)NPAIR * 4);
  float* pb     = (float*)alloc((size_t)NPAIR * 4);
  float* hA     = (float*)alloc((size_t)N_NODES * 64 * 4);
  float* hB     = (float*)alloc((size_t)N_NODES * 64 * 4);
  float* agg    = (float*)alloc((size_t)N_NODES * 64 * 4);
  float* h2     = (float*)alloc((size_t)N_NODES * 256 * 4);

  (void)hipMemsetAsync(cnt, 0, 128 * 4, stream);
  (void)hipMemsetAsync(degI, 0, N_NODES * 4, stream);
  basis_kernel<<<(N_EDGES + 255) / 256, 256, 0, stream>>>(pseudo, ei, wiA, bA, cnt, degI);
  scan_kernel<<<1, 1, 0, stream>>>(cnt, off, cursor);
  scatter_kernel<<<(NPAIR + 255) / 256, 256, 0, stream>>>(wiA, bA, pe, pb, cursor);
  dinv_kernel<<<(N_NODES + 255) / 256, 256, 0, stream>>>(degI, dinv);

  // layer 0: Fin=1 -> Fout=32
  (void)hipMemsetAsync(agg, 0, (size_t)N_NODES * 32 * 4, stream);
  l0_agg_kernel<<<((size_t)N_EDGES * 32 + 255) / 256, 256, 0, stream>>>(x, ei, wiA, bA, Wc[0], agg);
  finalize_kernel<<<N_NODES, 32, 0, stream>>>(agg, x, Rc[0], Bc[0], dinv, hA, 1);

  float* cur = hA; float* nxt = hB;
  int Fin = 32;
  for (int i = 1; i < 6; i++) {
    (void)hipMemsetAsync(agg, 0, (size_t)N_NODES * 64 * 4, stream);
    spline_agg_kernel<<<dim3(KTAB, 48), 32, 0, stream>>>(cur, ei, pe, pb, off, Wc[i], agg, Fin);
    finalize_kernel<<<N_NODES, 64, 0, stream>>>(agg, cur, Rc[i], Bc[i], dinv, nxt, Fin);
    float* tmp = cur; cur = nxt; nxt = tmp;
    Fin = 64;
  }

  fc1_kernel<<<dim3((N_NODES + 15) / 16, 4), 32, 0, stream>>>(cur, fc1w, fc1b, h2);
  fc2_kernel<<<dim3((N_NODES + 127) / 128, (N_NODES + 63) / 64), 256, 0, stream>>>(h2, fc2w, fc2b, out);
  logsoftmax_kernel<<<N_NODES, 256, 0, stream>>>(out);
}